// GCN_27006754357652
// MI455X (gfx1250) — compile-verified
//
#include <hip/hip_runtime.h>
#include <hip/hip_bf16.h>

// ---------------------------------------------------------------------------
// GCN forward, MI455X (gfx1250, wave32).
//   layer: h = relu( b + sum_{e:dst=i} dinv[src]*dinv[dst]*(xW)[src]
//                      + dinv[i]^2 * (xW)[i] )
// Dense GEMMs use V_WMMA_F32_16X16X4_F32 (fp32 in/out matches reference).
// Edge aggregation uses L2 float atomics (xW matrices fit in 192MB L2).
// ---------------------------------------------------------------------------

typedef float v2f __attribute__((ext_vector_type(2)));
typedef float v8f __attribute__((ext_vector_type(8)));

__global__ void fill_f32(float* __restrict__ p, int n, float v) {
    int i = blockIdx.x * blockDim.x + threadIdx.x;
    int stride = gridDim.x * blockDim.x;
    for (; i < n; i += stride) p[i] = v;
}

__global__ void deg_kernel(const int* __restrict__ dst, float* __restrict__ deg, int E) {
    int i = blockIdx.x * blockDim.x + threadIdx.x;
    if (i < E) atomicAdd(&deg[dst[i]], 1.0f);
}

__global__ void rsqrt_kernel(float* __restrict__ deg, int n) {
    int i = blockIdx.x * blockDim.x + threadIdx.x;
    if (i < n) deg[i] = rsqrtf(deg[i]);   // deg >= 1 always (self loops)
}

// C[M x Nd] = A[M x K] * B[K x Nd]; one 16x16 tile per wave.
// Requires M % 16 == 0, K % 4 == 0, Nd % 16 == 0.
// Fragment layouts per CDNA5 ISA 7.12.2 (f32 16x16x4):
//   A 16x4 : lane m=lane&15; vgpr0 = A[m][k+2*hi], vgpr1 = A[m][k+2*hi+1], hi=lane>>4
//   B 4x16 : lane n=lane&15; vgpr0 = B[k+2*hi][n], vgpr1 = B[k+2*hi+1][n]
//   D 16x16: vgpr r -> row r + 8*hi, col = lane&15
__global__ void gemm_wmma_f32(const float* __restrict__ A, const float* __restrict__ B,
                              float* __restrict__ C, int M, int K, int Nd) {
    int lane = threadIdx.x & 31;
    int wave = threadIdx.x >> 5;
    int tile = blockIdx.x * (blockDim.x >> 5) + wave;
    int tiles_n = Nd >> 4;
    int tm = tile / tiles_n;
    int tn = tile - tm * tiles_n;
    if (tm * 16 >= M) return;          // whole-wave exit -> EXEC all-1s for WMMA

    int m   = lane & 15;
    int hi  = lane >> 4;               // 0 or 1
    const float* Arow = A + (long)(tm * 16 + m) * K;
    int col = tn * 16 + m;

    v8f acc = {0.f, 0.f, 0.f, 0.f, 0.f, 0.f, 0.f, 0.f};
    for (int k = 0; k < K; k += 4) {
        int ka = k + 2 * hi;
        v2f a, b;
        a[0] = Arow[ka];
        a[1] = Arow[ka + 1];
        b[0] = B[(long)ka * Nd + col];
        b[1] = B[(long)(ka + 1) * Nd + col];
        acc = __builtin_amdgcn_wmma_f32_16x16x4_f32(
            /*neg_a=*/false, a, /*neg_b=*/false, b,
            /*c_mod=*/(short)0, acc, /*reuse_a=*/false, /*reuse_b=*/false);
    }

    float* Crow = C + (long)(tm * 16 + 8 * hi) * Nd + col;
#pragma unroll
    for (int r = 0; r < 8; ++r) Crow[(long)r * Nd] = acc[r];
}

// One thread per (edge, 4-float chunk): acc[dst] += dinv[src]*dinv[dst]*xw[src]
__global__ void scatter_kernel(const int* __restrict__ src, const int* __restrict__ dst,
                               const float* __restrict__ dinv, const float* __restrict__ xw,
                               float* __restrict__ acc, int E, int D) {
    int chunks = D >> 2;
    long total = (long)E * chunks;
    long i = blockIdx.x * (long)blockDim.x + threadIdx.x;
    long stride = (long)gridDim.x * blockDim.x;
    for (; i < total; i += stride) {
        int e = (int)(i / chunks);
        int c = (int)(i - (long)e * chunks) << 2;
        int s = src[e];
        int d = dst[e];
        float w = dinv[s] * dinv[d];
        const float4 v = *(const float4*)(xw + (long)s * D + c);
        float* o = acc + (long)d * D + c;
        atomicAdd(o + 0, w * v.x);
        atomicAdd(o + 1, w * v.y);
        atomicAdd(o + 2, w * v.z);
        atomicAdd(o + 3, w * v.w);
    }
}

// acc[i] = relu( acc[i] + dinv[node]^2 * xw[i] + bias[col] )
__global__ void finalize_kernel(float* __restrict__ acc, const float* __restrict__ xw,
                                const float* __restrict__ dinv, const float* __restrict__ bias,
                                int total, int D, int do_relu) {
    int i = blockIdx.x * blockDim.x + threadIdx.x;
    int stride = gridDim.x * blockDim.x;
    for (; i < total; i += stride) {
        int node = i / D;
        int col  = i - node * D;
        float di = dinv[node];
        float v = acc[i] + di * di * xw[i] + bias[col];
        acc[i] = do_relu ? fmaxf(v, 0.0f) : v;
    }
}

extern "C" void kernel_launch(void* const* d_in, const int* in_sizes, int n_in,
                              void* d_out, int out_size, void* d_ws, size_t ws_size,
                              hipStream_t stream) {
    const int IN = 32, HID = 64, OUT = 32;
    const float* x  = (const float*)d_in[0];
    const int* edge = (const int*)d_in[1];
    const float* W1 = (const float*)d_in[2];
    const float* b1 = (const float*)d_in[3];
    const float* W2 = (const float*)d_in[4];
    const float* b2 = (const float*)d_in[5];
    float* out = (float*)d_out;

    const int N = in_sizes[0] / IN;
    const int E = in_sizes[1] / 2;
    const int* srcv = edge;
    const int* dstv = edge + E;

    // workspace: dinv[N] | xw[N*HID] (reused as xw2) | acc1[N*HID]
    float* dinv = (float*)d_ws;
    float* xw   = dinv + N;
    float* acc1 = xw + (long)N * HID;

    const int T = 256;
    auto cdiv = [](long a, long b) { return (int)((a + b - 1) / b); };

    // ---- init ----
    fill_f32<<<cdiv((long)N * HID, T), T, 0, stream>>>(acc1, N * HID, 0.0f);
    fill_f32<<<cdiv(N, T), T, 0, stream>>>(dinv, N, 1.0f);          // self loop
    fill_f32<<<cdiv((long)N * OUT, T), T, 0, stream>>>(out, N * OUT, 0.0f);

    // ---- degrees -> dinv ----
    deg_kernel<<<cdiv(E, T), T, 0, stream>>>(dstv, dinv, E);
    rsqrt_kernel<<<cdiv(N, T), T, 0, stream>>>(dinv, N);

    // ---- layer 1 ----
    int tiles1 = (N / 16) * (HID / 16);
    gemm_wmma_f32<<<cdiv(tiles1, T / 32), T, 0, stream>>>(x, W1, xw, N, IN, HID);
    scatter_kernel<<<cdiv((long)E * (HID / 4), T), T, 0, stream>>>(
        srcv, dstv, dinv, xw, acc1, E, HID);
    finalize_kernel<<<cdiv((long)N * HID, T), T, 0, stream>>>(
        acc1, xw, dinv, b1, N * HID, HID, /*relu=*/1);

    // ---- layer 2 (accumulate straight into d_out) ----
    int tiles2 = (N / 16) * (OUT / 16);
    gemm_wmma_f32<<<cdiv(tiles2, T / 32), T, 0, stream>>>(acc1, W2, xw, N, HID, OUT);
    scatter_kernel<<<cdiv((long)E * (OUT / 4), T), T, 0, stream>>>(
        srcv, dstv, dinv, xw, out, E, OUT);
    finalize_kernel<<<cdiv((long)N * OUT, T), T, 0, stream>>>(
        out, xw, dinv, b2, N * OUT, OUT, /*relu=*/0);
}